// EulerAttentionHead_75892072121045
// MI455X (gfx1250) — compile-verified
//
#include <hip/hip_runtime.h>
#include <math.h>

#define LUT_SIZE 4096
#define B_SZ 32
#define T_SZ 4096
#define D_SZ 1024
#define KROW 130      // padded LDS row stride (floats) to avoid bank conflicts
#define TCHUNKS 64    // t-chunks for the bandwidth phase (64 t each)

typedef __attribute__((ext_vector_type(2))) float v2f;
typedef __attribute__((ext_vector_type(4))) float v4f;
typedef __attribute__((ext_vector_type(8))) float v8f;

__device__ __forceinline__ int lut_idx(float theta) {
    // idx = mod(floor(theta / TWO_PI * LUT_SIZE), LUT_SIZE); mask == python mod for pow2
    const float TWO_PI = 6.2831855f;  // np.float32(2*pi)
    int i = (int)floorf(theta / TWO_PI * 4096.0f);
    return i & (LUT_SIZE - 1);
}

// ---------------- Phase 1: query = [cos(theta_q), sin(theta_q)]  (32 x 128) ----------------
__global__ void __launch_bounds__(64) k_query(const float* __restrict__ x,
                                              const float* __restrict__ t,
                                              const float* __restrict__ wq,
                                              const float* __restrict__ bq,
                                              const float* __restrict__ sin_t,
                                              const float* __restrict__ cos_t,
                                              float* __restrict__ q_ws) {
    const float PHI = 1.6180340f;  // np.float32(golden ratio)
    int b = blockIdx.x;
    int j = threadIdx.x;  // 0..63
    float wl = 1.0f + fabsf(wq[j]);
    float theta = x[b * D_SZ + j] / wl + bq[j] + t[b] * PHI;
    int i = lut_idx(theta);
    q_ws[b * 128 + j] = cos_t[i];
    q_ws[b * 128 + 64 + j] = sin_t[i];
}

// ---------------- Phase 2: scores[b,t] = (q_b . key_{b,t}) / sqrt(128) via WMMA ----------------
// Each wave handles one 16-timestep tile: A = 16x4 key chunk (f32), B = query chunk
// broadcast over all 16 N columns. D columns are identical; lanes 0/16 write rows 0-7/8-15.
__global__ void __launch_bounds__(128) k_scores(const float* __restrict__ cached,
                                                const float* __restrict__ wk,
                                                const float* __restrict__ bk,
                                                const float* __restrict__ sin_t,
                                                const float* __restrict__ cos_t,
                                                const float* __restrict__ q_ws,
                                                float* __restrict__ scores) {
    __shared__ float ldsK[4][16 * KROW];  // per-wave 16x128 key tile (padded rows)
    __shared__ float ldsQ[128];

    int b    = blockIdx.y;
    int wave = threadIdx.x >> 5;
    int lane = threadIdx.x & 31;
    int t0   = (blockIdx.x * 4 + wave) * 16;

    ldsQ[threadIdx.x] = q_ws[b * 128 + threadIdx.x];  // blockDim == 128, no divergence

    // Build keys tile via LUT: key[t][d] = d<64 ? cos(theta_k[d]) : sin(theta_k[d-64])
    const float* cbase = cached + ((size_t)b * T_SZ + t0) * D_SZ;
    for (int idx = lane; idx < 16 * 64; idx += 32) {
        int m = idx >> 6;
        int j = idx & 63;
        float wl = 1.0f + fabsf(wk[j]);
        float theta = cbase[(size_t)m * D_SZ + j] / wl + bk[j];
        int i = lut_idx(theta);
        ldsK[wave][m * KROW + j]      = cos_t[i];
        ldsK[wave][m * KROW + 64 + j] = sin_t[i];
    }
    __syncthreads();

    // WMMA f32 16x16x4 layouts:
    //   A: lanes 0-15 hold M=lane with K=0 (v0), K=1 (v1); lanes 16-31 hold K=2,K=3
    //   B: lanes 0-15 hold rows K=0 (v0), K=1 (v1); lanes 16-31 rows K=2, K=3
    int half = lane >> 4;
    int m    = lane & 15;
    const float* rowA = &ldsK[wave][m * KROW];
    v8f c = {0.f, 0.f, 0.f, 0.f, 0.f, 0.f, 0.f, 0.f};
#pragma unroll
    for (int k = 0; k < 128; k += 4) {
        int kk = k + 2 * half;
        v2f a;  a[0]  = rowA[kk];  a[1]  = rowA[kk + 1];
        v2f bb; bb[0] = ldsQ[kk];  bb[1] = ldsQ[kk + 1];
        c = __builtin_amdgcn_wmma_f32_16x16x4_f32(false, a, false, bb,
                                                  (short)0, c, false, false);
    }

    const float invScale = 0.08838834764831845f;  // 1/sqrt(2*64)
    if (m == 0) {  // lane 0 -> rows 0..7, lane 16 -> rows 8..15 (all N columns equal)
        int tb = t0 + half * 8;
#pragma unroll
        for (int i2 = 0; i2 < 8; ++i2)
            scores[b * T_SZ + tb + i2] = c[i2] * invScale;
    }
}

// ---------------- Phase 3: row softmax over T=4096, weights -> output segment 2 ----------------
__global__ void __launch_bounds__(256) k_softmax(const float* __restrict__ scores,
                                                 float* __restrict__ weights) {
    __shared__ float red[256];
    int b = blockIdx.x;
    int tid = threadIdx.x;
    const float* s = scores + b * T_SZ;
    float* w = weights + b * T_SZ;

    float mx = -INFINITY;
    for (int i = tid; i < T_SZ; i += 256) mx = fmaxf(mx, s[i]);
    red[tid] = mx;
    __syncthreads();
    for (int off = 128; off > 0; off >>= 1) {
        if (tid < off) red[tid] = fmaxf(red[tid], red[tid + off]);
        __syncthreads();
    }
    mx = red[0];
    __syncthreads();

    float sum = 0.f;
    for (int i = tid; i < T_SZ; i += 256) {
        float e = expf(s[i] - mx);
        w[i] = e;
        sum += e;
    }
    red[tid] = sum;
    __syncthreads();
    for (int off = 128; off > 0; off >>= 1) {
        if (tid < off) red[tid] += red[tid + off];
        __syncthreads();
    }
    float inv = 1.0f / red[0];
    for (int i = tid; i < T_SZ; i += 256) w[i] *= inv;
}

// ---------------- Phase 4: partial[b,tc,:] = sum_{t in chunk} w[b,t]*cached[b,t,:] ----------------
// Bandwidth phase: streams all 512 MB of cached_states once with b128 non-temporal loads.
// One block covers a full 1024-float row with float4 lanes; 64 t per chunk, 2048 blocks.
__global__ void __launch_bounds__(256) k_partial(const float* __restrict__ cached,
                                                 const float* __restrict__ weights,
                                                 float* __restrict__ partial) {
    __shared__ float wloc[64];
    int tc  = blockIdx.x;  // 0..TCHUNKS-1
    int b   = blockIdx.y;  // 0..31
    int tid = threadIdx.x; // float4 lane: covers d = 4*tid .. 4*tid+3
    int t0 = tc * (T_SZ / TCHUNKS);
    if (tid < (T_SZ / TCHUNKS)) wloc[tid] = weights[b * T_SZ + t0 + tid];
    __syncthreads();

    const v4f* cb = (const v4f*)(cached + ((size_t)b * T_SZ + t0) * D_SZ) + tid;
    v4f acc = {0.f, 0.f, 0.f, 0.f};
#pragma unroll 4
    for (int tt = 0; tt < (T_SZ / TCHUNKS); ++tt) {
        v4f v = __builtin_nontemporal_load(cb + (size_t)tt * (D_SZ / 4));
        float w = wloc[tt];
        acc.x = fmaf(w, v.x, acc.x);
        acc.y = fmaf(w, v.y, acc.y);
        acc.z = fmaf(w, v.z, acc.z);
        acc.w = fmaf(w, v.w, acc.w);
    }
    v4f* p = (v4f*)(partial + ((size_t)(b * TCHUNKS + tc)) * D_SZ) + tid;
    *p = acc;
}

// ---------------- Phase 5: deterministic reduction over TCHUNKS t-chunks (b128) ----------------
__global__ void __launch_bounds__(256) k_reduce(const float* __restrict__ partial,
                                                float* __restrict__ out) {
    int b   = blockIdx.x;  // 0..31
    int tid = threadIdx.x; // float4 lane over the 1024-wide row
    const v4f* p = (const v4f*)(partial + ((size_t)b * TCHUNKS) * D_SZ) + tid;
    v4f acc = {0.f, 0.f, 0.f, 0.f};
#pragma unroll
    for (int c = 0; c < TCHUNKS; ++c) {
        v4f v = p[(size_t)c * (D_SZ / 4)];
        acc.x += v.x; acc.y += v.y; acc.z += v.z; acc.w += v.w;
    }
    ((v4f*)(out + (size_t)b * D_SZ))[tid] = acc;
}

extern "C" void kernel_launch(void* const* d_in, const int* in_sizes, int n_in,
                              void* d_out, int out_size, void* d_ws, size_t ws_size,
                              hipStream_t stream) {
    const float* x      = (const float*)d_in[0];
    const float* cached = (const float*)d_in[1];
    const float* t      = (const float*)d_in[2];
    const float* wq     = (const float*)d_in[3];
    const float* bq     = (const float*)d_in[4];
    const float* wk     = (const float*)d_in[5];
    const float* bk     = (const float*)d_in[6];
    const float* sin_t  = (const float*)d_in[7];
    const float* cos_t  = (const float*)d_in[8];

    float* out_o = (float*)d_out;            // output: (32, 1024)
    float* out_w = out_o + B_SZ * D_SZ;      // weights: (32, 4096)

    float* ws      = (float*)d_ws;
    float* q_ws    = ws;                     // 32*128   = 16 KB
    float* scores  = q_ws + B_SZ * 128;      // 32*4096  = 512 KB
    float* partial = scores + B_SZ * T_SZ;   // 32*64*1024 = 8 MB

    k_query<<<dim3(B_SZ), dim3(64), 0, stream>>>(x, t, wq, bq, sin_t, cos_t, q_ws);
    k_scores<<<dim3(64, B_SZ), dim3(128), 0, stream>>>(cached, wk, bk, sin_t, cos_t,
                                                       q_ws, scores);
    k_softmax<<<dim3(B_SZ), dim3(256), 0, stream>>>(scores, out_w);
    k_partial<<<dim3(TCHUNKS, B_SZ), dim3(256), 0, stream>>>(cached, out_w, partial);
    k_reduce<<<dim3(B_SZ), dim3(256), 0, stream>>>(partial, out_o);
}